// QuantumCognitionModel_58420145160763
// MI455X (gfx1250) — compile-verified
//
#include <hip/hip_runtime.h>
#include <hip/hip_bf16.h>

typedef __attribute__((ext_vector_type(2))) float v2f;
typedef __attribute__((ext_vector_type(8))) float v8f;

#define D_ 32
#define N_ 64
#define B_ 4096
#define NB 16          // batch columns per solver block
#define PRE_IT 12      // power iterations on H (PSD) to estimate lambda_max
#define MAIN_IT 84     // shifted power iterations on cI - H -> lowest eigvec

// ---- f32 WMMA 16x16x4 (CDNA5): D = A(16x4) * B(4x16) + C(16x16 f32) ----
#if __has_builtin(__builtin_amdgcn_wmma_f32_16x16x4_f32)
__device__ __forceinline__ v8f wmma4(v2f a, v2f b, v8f c) {
  return __builtin_amdgcn_wmma_f32_16x16x4_f32(false, a, false, b, (short)0, c,
                                               false, false);
}
#else
__device__ __forceinline__ v8f wmma4(v2f a, v2f b, v8f c) {
  v8f d;
  asm volatile("v_wmma_f32_16x16x4_f32 %0, %1, %2, %3"
               : "=v"(d)
               : "v"(a), "v"(b), "v"(c));
  return d;
}
#endif

// ---------------- Kernel 1: A = 0.5*(W + W^H), split re/im planes ----------
__global__ __launch_bounds__(256) void k_hermitize(const float* __restrict__ W,
                                                   float* __restrict__ Ar,
                                                   float* __restrict__ Ai) {
  const int k = blockIdx.x;  // 0..31
  const float* Wk = W + k * (N_ * N_ * 2);
  float* ArK = Ar + k * N_ * N_;
  float* AiK = Ai + k * N_ * N_;
  for (int idx = threadIdx.x; idx < N_ * N_; idx += 256) {
    int i = idx >> 6, j = idx & 63;
    float wr_ij = Wk[(i * N_ + j) * 2 + 0], wi_ij = Wk[(i * N_ + j) * 2 + 1];
    float wr_ji = Wk[(j * N_ + i) * 2 + 0], wi_ji = Wk[(j * N_ + i) * 2 + 1];
    ArK[idx] = 0.5f * (wr_ij + wr_ji);
    AiK[idx] = 0.5f * (wi_ij - wi_ji);
  }
}

// ---------------- Kernel 2: S2 = sum_k A_k * A_k (complex, 64x64) ----------
__global__ __launch_bounds__(256) void k_sumA2(const float* __restrict__ Ar,
                                               const float* __restrict__ Ai,
                                               float* __restrict__ S2r,
                                               float* __restrict__ S2i) {
  const int g = blockIdx.x * 256 + threadIdx.x;  // 0..4095
  const int i = g >> 6, j = g & 63;
  float sr = 0.f, si = 0.f;
  for (int k = 0; k < D_; ++k) {
    const float* R = Ar + k * N_ * N_;
    const float* I = Ai + k * N_ * N_;
    for (int m = 0; m < N_; ++m) {
      float a = R[i * N_ + m], b = I[i * N_ + m];
      float c = R[m * N_ + j], d = I[m * N_ + j];
      sr = fmaf(a, c, fmaf(-b, d, sr));
      si = fmaf(a, d, fmaf(b, c, si));
    }
  }
  S2r[g] = sr;
  S2i[g] = si;
}

// Complex 16x16 tile product accumulator: T = M * V for one matrix row-tile.
// arf/aif: per-lane A-row fragments (16 k-steps); brf/bif: B fragments.
// Results: Tr = Prr - Pii, Ti = Pri + Pir (computed by caller from products).
#define CPLX_WMMA_CHAIN(arf, aif, brf, bif, Prr, Pii, Pri, Pir)   \
  _Pragma("unroll") for (int ks = 0; ks < 16; ++ks) {             \
    Prr = wmma4(arf[ks], brf[ks], Prr);                           \
    Pii = wmma4(aif[ks], bif[ks], Pii);                           \
    Pri = wmma4(arf[ks], bif[ks], Pri);                           \
    Pir = wmma4(aif[ks], brf[ks], Pir);                           \
  }

// ---------------- Kernel 3: batched bottom-eigenvector solver --------------
// One block handles NB=16 batch columns. V is 64x16 complex in LDS.
// H*V = 0.5*S2*V + sum_k(-x_bk)*(A_k*V) + 0.5*x2*V, all matmuls via WMMA f32.
__global__ __launch_bounds__(256) void k_solver(const float* __restrict__ Ar,
                                                const float* __restrict__ Ai,
                                                const float* __restrict__ S2r,
                                                const float* __restrict__ S2i,
                                                const float* __restrict__ x,
                                                float* __restrict__ out) {
  __shared__ float sVr[N_ * NB], sVi[N_ * NB];   // current iterate (64x16)
  __shared__ float sUr[N_ * NB], sUi[N_ * NB];   // H*V, then u, scratch
  __shared__ float sPR[2 * 4 * 16 * 16];         // [kh][mt][row16][col]
  __shared__ float sPI[2 * 4 * 16 * 16];
  __shared__ float sXs[D_ * NB];                 // x transposed: [k][col]
  __shared__ float sX2[NB], sC[NB];
  __shared__ float sNrmP[4 * 16];                // row-quarter norm partials
  __shared__ float sRhoP[4 * 16];                // Rayleigh partials
  __shared__ float sPosR[2 * D_ * NB], sPosI[2 * D_ * NB], sE2[2 * D_ * NB];

  const int tid = threadIdx.x;
  const int lane = tid & 31;
  const int wid = tid >> 5;  // 0..7
  const int col = lane & 15;
  const int b0 = blockIdx.x * NB;

  // stage x (transposed) and deterministic pseudo-random init of V
  for (int idx = tid; idx < D_ * NB; idx += 256) {
    int k = idx >> 4, c = idx & 15;
    sXs[idx] = x[(b0 + c) * D_ + k];
  }
  for (int idx = tid; idx < N_ * NB; idx += 256) {
    unsigned h = (unsigned)(idx + 1025u * blockIdx.x + 1u) * 2654435761u;
    h ^= h >> 16; h *= 2246822519u; h ^= h >> 13;
    sVr[idx] = (float)(h & 0xFFFFu) * (1.f / 65536.f) - 0.5f;
    h *= 2654435761u; h ^= h >> 16;
    sVi[idx] = (float)(h & 0xFFFFu) * (1.f / 65536.f) - 0.5f;
  }
  __syncthreads();
  if (tid < NB) {
    float s = 0.f;
    for (int k = 0; k < D_; ++k) { float v = sXs[k * NB + tid]; s += v * v; }
    sX2[tid] = s;
    sC[tid] = 0.f;
  }
  __syncthreads();

  const int mt = wid & 3;                 // output M-tile (rows mt*16..+15)
  const int kh = wid >> 2;                // matrix-half this wave sums
  const int mstart = kh ? 17 : 0;
  const int mend = kh ? 33 : 17;          // m==32 is the S2 term
  const int rowA = mt * 16 + col;         // A row supplied by this lane
  const int koff = (lane < 16) ? 0 : 2;   // K sub-offset per half-wave

  for (int it = 0; it < PRE_IT + MAIN_IT; ++it) {
    // hoist B (V-column) fragments into registers once per iteration
    v2f brf[16], bif[16];
#pragma unroll
    for (int ks = 0; ks < 16; ++ks) {
      const int kb = ks * 4 + koff;
      brf[ks].x = sVr[(kb + 0) * NB + col];
      brf[ks].y = sVr[(kb + 1) * NB + col];
      bif[ks].x = sVi[(kb + 0) * NB + col];
      bif[ks].y = sVi[(kb + 1) * NB + col];
    }

    v8f accR = {}, accI = {};
    for (int m = mstart; m < mend; ++m) {
      const float* Mr = (m < D_) ? (Ar + m * N_ * N_) : S2r;
      const float* Mi = (m < D_) ? (Ai + m * N_ * N_) : S2i;
      const float w = (m < D_) ? -sXs[m * NB + col] : 0.5f;

      // burst-load this matrix's per-lane A-row fragments (one wait point)
      v2f arf[16], aif[16];
#pragma unroll
      for (int ks = 0; ks < 16; ++ks)
        arf[ks] = *(const v2f*)(Mr + rowA * N_ + ks * 4 + koff);
#pragma unroll
      for (int ks = 0; ks < 16; ++ks)
        aif[ks] = *(const v2f*)(Mi + rowA * N_ + ks * 4 + koff);

      // prefetch next matrix's rows (near scope) under this matrix's WMMAs
      if (m + 1 < mend) {
        const float* Nr = (m + 1 < D_) ? (Ar + (m + 1) * N_ * N_) : S2r;
        const float* Ni = (m + 1 < D_) ? (Ai + (m + 1) * N_ * N_) : S2i;
        __builtin_prefetch(Nr + rowA * N_, 0, 3);
        __builtin_prefetch(Ni + rowA * N_, 0, 3);
      }

      v8f Prr = {}, Pii = {}, Pri = {}, Pir = {};
      CPLX_WMMA_CHAIN(arf, aif, brf, bif, Prr, Pii, Pri, Pir)

#pragma unroll
      for (int e = 0; e < 8; ++e) {
        accR[e] = fmaf(w, Prr[e] - Pii[e], accR[e]);
        accI[e] = fmaf(w, Pri[e] + Pir[e], accI[e]);
      }
    }
    {
      const int rbase = (kh * 4 + mt) * 16;
#pragma unroll
      for (int e = 0; e < 8; ++e) {
        int r = e + ((lane < 16) ? 0 : 8);
        sPR[(rbase + r) * 16 + col] = accR[e];
        sPI[(rbase + r) * 16 + col] = accI[e];
      }
    }
    __syncthreads();

    // h = H*V (combine the two k-half partials + diagonal term), store in sU
    for (int idx = tid; idx < N_ * NB; idx += 256) {
      int row = idx >> 4, c = idx & 15;
      int m2 = row >> 4, r = row & 15;
      sUr[idx] = sPR[((0 * 4 + m2) * 16 + r) * 16 + c] +
                 sPR[((1 * 4 + m2) * 16 + r) * 16 + c] +
                 0.5f * sX2[c] * sVr[idx];
      sUi[idx] = sPI[((0 * 4 + m2) * 16 + r) * 16 + c] +
                 sPI[((1 * 4 + m2) * 16 + r) * 16 + c] +
                 0.5f * sX2[c] * sVi[idx];
    }
    __syncthreads();

    if (it == PRE_IT) {  // Rayleigh quotient -> shift c (deterministic tree)
      if (tid < 64) {
        int c = tid & 15, q = tid >> 4;
        float s = 0.f;
        for (int r = q * 16; r < q * 16 + 16; ++r) {
          int idx = r * NB + c;
          s = fmaf(sVr[idx], sUr[idx], fmaf(sVi[idx], sUi[idx], s));
        }
        sRhoP[q * 16 + c] = s;
      }
      __syncthreads();
      if (tid < NB) {
        float rho = sRhoP[0 * 16 + tid] + sRhoP[1 * 16 + tid] +
                    sRhoP[2 * 16 + tid] + sRhoP[3 * 16 + tid];
        sC[tid] = rho * 1.02f + 1e-3f;
      }
      __syncthreads();
    }

    // u = h (pre phase, dominant eig of PSD H) or c*v - h (main phase)
    for (int idx = tid; idx < N_ * NB; idx += 256) {
      int c = idx & 15;
      float ur, ui;
      if (it < PRE_IT) { ur = sUr[idx]; ui = sUi[idx]; }
      else {
        ur = sC[c] * sVr[idx] - sUr[idx];
        ui = sC[c] * sVi[idx] - sUi[idx];
      }
      sUr[idx] = ur; sUi[idx] = ui;   // own-index in-place, race-free
    }
    __syncthreads();
    if (tid < 64) {  // deterministic column norms
      int c = tid & 15, q = tid >> 4;
      float s = 0.f;
      for (int r = q * 16; r < q * 16 + 16; ++r) {
        int idx = r * NB + c;
        s = fmaf(sUr[idx], sUr[idx], fmaf(sUi[idx], sUi[idx], s));
      }
      sNrmP[q * 16 + c] = s;
    }
    __syncthreads();
    for (int idx = tid; idx < N_ * NB; idx += 256) {
      int c = idx & 15;
      float nrm = sNrmP[0 * 16 + c] + sNrmP[1 * 16 + c] + sNrmP[2 * 16 + c] +
                  sNrmP[3 * 16 + c];
      float s = rsqrtf(nrm + 1e-30f);
      sVr[idx] = sUr[idx] * s;
      sVi[idx] = sUi[idx] * s;
    }
    __syncthreads();
  }

  // ---- outputs: y = A_k*phi via WMMA; pos=phi^H y, exp2=|y|^2 ----
  // wave wid handles k = wid*4..wid*4+3; lane halves write disjoint slots.
  {
    v2f brf[16], bif[16];
#pragma unroll
    for (int ks = 0; ks < 16; ++ks) {
      const int kb = ks * 4 + koff;
      brf[ks].x = sVr[(kb + 0) * NB + col];
      brf[ks].y = sVr[(kb + 1) * NB + col];
      bif[ks].x = sVi[(kb + 0) * NB + col];
      bif[ks].y = sVi[(kb + 1) * NB + col];
    }
    for (int mm = 0; mm < 4; ++mm) {
      const int m = wid * 4 + mm;
      const float* Mr = Ar + m * N_ * N_;
      const float* Mi = Ai + m * N_ * N_;
      float pr = 0.f, pi = 0.f, ee = 0.f;
      for (int m2 = 0; m2 < 4; ++m2) {
        const int row0 = m2 * 16 + col;
        v2f arf[16], aif[16];
#pragma unroll
        for (int ks = 0; ks < 16; ++ks)
          arf[ks] = *(const v2f*)(Mr + row0 * N_ + ks * 4 + koff);
#pragma unroll
        for (int ks = 0; ks < 16; ++ks)
          aif[ks] = *(const v2f*)(Mi + row0 * N_ + ks * 4 + koff);

        v8f Prr = {}, Pii = {}, Pri = {}, Pir = {};
        CPLX_WMMA_CHAIN(arf, aif, brf, bif, Prr, Pii, Pri, Pir)

#pragma unroll
        for (int e = 0; e < 8; ++e) {
          float tr = Prr[e] - Pii[e];
          float ti = Pri[e] + Pir[e];
          int row = m2 * 16 + e + ((lane < 16) ? 0 : 8);
          float fr = sVr[row * NB + col];
          float fi = sVi[row * NB + col];
          pr = fmaf(fr, tr, fmaf(fi, ti, pr));      // Re(conj(phi).y)
          pi = fmaf(fr, ti, fmaf(-fi, tr, pi));     // Im(conj(phi).y)
          ee = fmaf(tr, tr, fmaf(ti, ti, ee));      // |y|^2
        }
      }
      const int half = (lane < 16) ? 0 : 1;
      sPosR[(half * D_ + m) * NB + col] = pr;
      sPosI[(half * D_ + m) * NB + col] = pi;
      sE2[(half * D_ + m) * NB + col] = ee;
    }
  }
  __syncthreads();

  for (int idx = tid; idx < D_ * NB; idx += 256) {
    int m = idx >> 4, c = idx & 15;
    float prr = sPosR[idx] + sPosR[D_ * NB + idx];
    float pii = sPosI[idx] + sPosI[D_ * NB + idx];
    out[((b0 + c) * D_ + m) * 2 + 0] = prr;
    out[((b0 + c) * D_ + m) * 2 + 1] = pii;
  }
  __syncthreads();
  if (tid < NB) {
    float wr = 0.f, wi = 0.f;
    for (int m = 0; m < D_; ++m) {
      float prr = sPosR[m * NB + tid] + sPosR[(D_ + m) * NB + tid];
      float pii = sPosI[m * NB + tid] + sPosI[(D_ + m) * NB + tid];
      float e2 = sE2[m * NB + tid] + sE2[(D_ + m) * NB + tid];
      wr += e2 - (prr * prr - pii * pii);
      wi += -2.f * prr * pii;
    }
    float* wv = out + B_ * D_ * 2;
    wv[(b0 + tid) * 2 + 0] = 0.05f * wr;
    wv[(b0 + tid) * 2 + 1] = 0.05f * wi;
  }
}

extern "C" void kernel_launch(void* const* d_in, const int* in_sizes, int n_in,
                              void* d_out, int out_size, void* d_ws,
                              size_t ws_size, hipStream_t stream) {
  const float* x = (const float*)d_in[0];   // (B, D) float32
  const float* W = (const float*)d_in[1];   // (D, N, N) complex64 interleaved
  float* ws = (float*)d_ws;
  float* Ar = ws;                       // D*N*N
  float* Ai = Ar + D_ * N_ * N_;        // D*N*N
  float* S2r = Ai + D_ * N_ * N_;       // N*N
  float* S2i = S2r + N_ * N_;           // N*N
  float* out = (float*)d_out;           // pos (B*D complex) then wvar (B complex)

  k_hermitize<<<D_, 256, 0, stream>>>(W, Ar, Ai);
  k_sumA2<<<(N_ * N_) / 256, 256, 0, stream>>>(Ar, Ai, S2r, S2i);
  k_solver<<<B_ / NB, 256, 0, stream>>>(Ar, Ai, S2r, S2i, x, out);
}